// GroupPool_2869038153934
// MI455X (gfx1250) — compile-verified
//
#include <hip/hip_runtime.h>

typedef float v2f __attribute__((ext_vector_type(2)));
typedef float v4f __attribute__((ext_vector_type(4)));
typedef float v8f __attribute__((ext_vector_type(8)));

#define D_FEAT 64   // feature dim (reference: D = 64)

// ---------------------------------------------------------------------------
// Kernel 0: zero the accumulators (d_out = sums, d_ws = counts). Harness
// poisons buffers, so we must re-zero on every call (graph replay safe).
// ---------------------------------------------------------------------------
__global__ void gp_zero(float* __restrict__ out, float* __restrict__ counts,
                        int out_n, int g_n) {
    int stride = gridDim.x * blockDim.x;
    int total  = out_n + g_n;
    for (int i = blockIdx.x * blockDim.x + threadIdx.x; i < total; i += stride) {
        if (i < out_n) out[i] = 0.0f;
        else           counts[i - out_n] = 0.0f;
    }
}

// ---------------------------------------------------------------------------
// Kernel 1: scatter-accumulate. 16 lanes per row; each lane issues one
// non-temporal 128-bit load (x streamed once -> NT keeps the 4MB accumulator
// table resident in the 192MB L2) and 4 relaxed agent-scope f32 atomic adds
// (lower to non-returning global_atomic_add_f32 serviced in L2).
// ---------------------------------------------------------------------------
__global__ void gp_accum(const v4f* __restrict__ x4,
                         const int* __restrict__ group,
                         float* __restrict__ sums,
                         float* __restrict__ counts,
                         int n_rows, int n_groups) {
    long long gtid = (long long)blockIdx.x * blockDim.x + threadIdx.x;
    int row = (int)(gtid >> 4);
    int sub = (int)(gtid & 15);
    if (row >= n_rows) return;

    int g = group[row];
    if ((unsigned)g >= (unsigned)n_groups) return;  // safety

    v4f v = __builtin_nontemporal_load(&x4[(size_t)row * (D_FEAT / 4) + sub]);

    float* dst = sums + (size_t)g * D_FEAT + sub * 4;
    __hip_atomic_fetch_add(&dst[0], v.x, __ATOMIC_RELAXED, __HIP_MEMORY_SCOPE_AGENT);
    __hip_atomic_fetch_add(&dst[1], v.y, __ATOMIC_RELAXED, __HIP_MEMORY_SCOPE_AGENT);
    __hip_atomic_fetch_add(&dst[2], v.z, __ATOMIC_RELAXED, __HIP_MEMORY_SCOPE_AGENT);
    __hip_atomic_fetch_add(&dst[3], v.w, __ATOMIC_RELAXED, __HIP_MEMORY_SCOPE_AGENT);

    if (sub == 0)
        __hip_atomic_fetch_add(&counts[g], 1.0f, __ATOMIC_RELAXED, __HIP_MEMORY_SCOPE_AGENT);
}

// ---------------------------------------------------------------------------
// Kernel 2: finalize mean via WMMA. Per 16x16 tile (16 groups x 16 features):
//   out_tile = diag(1/max(count,1)) @ sums_tile
// computed as 4 chained V_WMMA_F32_16X16X4_F32 (K = 16 split into 4x4).
// Exact: each output element is a single f32 multiply (off-diagonal terms 0).
// A 16x4 layout: lane holds row M=lane%16; VGPR0/1 = K {0,1} (lanes<16) or
// {2,3} (lanes>=16). B 4x16: lane holds col N=lane%16; VGPR0/1 = K {0,1} /
// {2,3} by lane half. D: VGPR j, lane -> (M = j + 8*half, N = lane%16).
// ---------------------------------------------------------------------------
__global__ void gp_finalize(float* __restrict__ out,
                            const float* __restrict__ counts,
                            int n_groups) {
    const int dtiles = D_FEAT / 16;                       // 4
    int ntiles = (n_groups / 16) * dtiles;                // tiles of 16g x 16d
    int wave = (int)((blockIdx.x * blockDim.x + threadIdx.x) >> 5);
    if (wave >= ntiles) return;                           // wave-uniform: EXEC all-ones

    int lane = threadIdx.x & 31;
    int gt = wave / dtiles, dt = wave % dtiles;
    int g_base = gt * 16, d_base = dt * 16;
    int mm   = lane & 15;      // row M (for A) and col N (for B/D)
    int half = lane >> 4;

    float cnt = counts[g_base + mm];
    float inv = 1.0f / fmaxf(cnt, 1.0f);

    v8f c = {};
#pragma unroll
    for (int k = 0; k < 4; ++k) {
        int k0 = 4 * k + 2 * half;   // global K of this lane-half's VGPR0 slot
        v2f a, b;
        a.x = (mm == k0)     ? inv : 0.0f;   // A[M][k0]
        a.y = (mm == k0 + 1) ? inv : 0.0f;   // A[M][k0+1]
        b.x = out[(size_t)(g_base + k0)     * D_FEAT + d_base + mm]; // B[k0][N]
        b.y = out[(size_t)(g_base + k0 + 1) * D_FEAT + d_base + mm]; // B[k0+1][N]
        c = __builtin_amdgcn_wmma_f32_16x16x4_f32(
                /*neg_a=*/false, a, /*neg_b=*/false, b,
                /*c_mod=*/(short)0, c, /*reuse_a=*/false, /*reuse_b=*/false);
    }

#pragma unroll
    for (int j = 0; j < 8; ++j)
        out[(size_t)(g_base + j + 8 * half) * D_FEAT + d_base + mm] = c[j];
}

// ---------------------------------------------------------------------------
extern "C" void kernel_launch(void* const* d_in, const int* in_sizes, int n_in,
                              void* d_out, int out_size, void* d_ws, size_t ws_size,
                              hipStream_t stream) {
    const float* x     = (const float*)d_in[0];
    const int*   group = (const int*)d_in[1];
    (void)d_in[2]; (void)n_in; (void)ws_size;

    int N = in_sizes[1];            // rows
    int G = out_size / D_FEAT;      // groups

    float* out    = (float*)d_out;  // sums accumulate in place, then scaled
    float* counts = (float*)d_ws;   // G floats of scratch

    // 0) zero sums + counts
    {
        int total = out_size + G;
        int blk = 256;
        int grd = (total + blk - 1) / blk;
        gp_zero<<<grd, blk, 0, stream>>>(out, counts, out_size, G);
    }

    // 1) scatter-accumulate: 16 lanes per row
    {
        long long threads = (long long)N * 16;
        int blk = 256;
        unsigned grd = (unsigned)((threads + blk - 1) / blk);
        gp_accum<<<grd, blk, 0, stream>>>((const v4f*)x, group, out, counts, N, G);
    }

    // 2) WMMA finalize: one wave per 16x16 tile
    {
        int ntiles = (G / 16) * (D_FEAT / 16);
        int blk = 256;                         // 8 waves/block
        int grd = (ntiles + 7) / 8;
        gp_finalize<<<grd, blk, 0, stream>>>(out, counts, G);
    }
}